// GateGCNLayer_71055938945247
// MI455X (gfx1250) — compile-verified
//
#include <hip/hip_runtime.h>
#include <hip/hip_bf16.h>
#include <math.h>

// ---------------- problem constants ----------------
#define NN 50000
#define NE 600000
#define DD 128
#define GNODE 5000
#define GEDGE 60000
#define EPS_NORM 1e-5f
#define EPS_RED  1e-6f
#define ND ((size_t)NN * DD)   // 6,400,000
#define ED ((size_t)NE * DD)   // 76,800,000

#define NTILE 80    // node rows per block  (50000 = 625 * 80), 5 row-tiles
#define ETILE 64    // edges per block      (600000 = 9375 * 64), 4 row-tiles

typedef float v2f __attribute__((ext_vector_type(2)));
typedef float v8f __attribute__((ext_vector_type(8)));
typedef int   v4i __attribute__((vector_size(16)));

#define GLOBAL_AS __attribute__((address_space(1)))
#define LDS_AS    __attribute__((address_space(3)))

#if defined(__has_builtin)
#  if __has_builtin(__builtin_amdgcn_global_load_async_to_lds_b128) && \
      __has_builtin(__builtin_amdgcn_s_wait_asynccnt)
#    define USE_ASYNC_LDS 1
#  endif
#endif
#ifndef USE_ASYNC_LDS
#  define USE_ASYNC_LDS 0
#endif

// =====================================================================
// Kernel 1: node GEMMs  Ah,Bh,Dh,Eh = h @ W{0,1,3,4} + b{0,1,3,4}
// One block = 80 rows (5 row-tiles). 8 waves: wave -> (weight, col-half).
// fp32 WMMA 16x16x4; B fragment reused across 5 row-tiles.
// =====================================================================
__global__ __launch_bounds__(256) void node_gemm_kernel(
    const float* __restrict__ h, const float* __restrict__ W,
    const float* __restrict__ b,
    float* __restrict__ Ah, float* __restrict__ Bh,
    float* __restrict__ Dh, float* __restrict__ Eh)
{
    __shared__ float tile[NTILE][132];         // padded: bank-conflict-free A reads
    const int t = threadIdx.x;
    const size_t row0 = (size_t)blockIdx.x * NTILE;

    // stage 80x128 floats = 2560 float4, 10 per thread
    {
        const float* gbase = h + row0 * DD;
#if USE_ASYNC_LDS
        #pragma unroll
        for (int i = 0; i < 10; ++i) {
            int q = t + 256 * i;
            int row = q >> 5;
            int c4  = q & 31;
            __builtin_amdgcn_global_load_async_to_lds_b128(
                (GLOBAL_AS v4i*)(gbase + (size_t)row * DD + c4 * 4),
                (LDS_AS v4i*)&tile[row][c4 * 4], 0, 0);
        }
        __builtin_amdgcn_s_wait_asynccnt(0);
#else
        const float4* src = (const float4*)gbase;
        #pragma unroll
        for (int i = 0; i < 10; ++i) {
            int q = t + 256 * i;
            int row = q >> 5;
            int c4  = q & 31;
            float4 v = src[row * 32 + c4];
            *(float4*)&tile[row][c4 * 4] = v;
        }
#endif
    }
    __syncthreads();

    const int wave = t >> 5;
    const int lane = t & 31;
    const int weight = wave >> 1;                       // 0..3 -> Ah,Bh,Dh,Eh
    const int half   = wave & 1;                        // col tiles [half*4, half*4+4)
    const int widx   = weight + (weight >= 2 ? 1 : 0);  // W index 0,1,3,4
    const float* __restrict__ Wm = W + (size_t)widx * DD * DD;
    const float* __restrict__ bv = b + (size_t)widx * DD;
    float* __restrict__ outp = (weight == 0) ? Ah : (weight == 1) ? Bh
                              : (weight == 2) ? Dh : Eh;

    const int m  = lane & 15;    // A row / B,C column-in-tile
    const int kg = lane >> 4;    // K sub-group (0 or 1)

    for (int ct = half * 4; ct < half * 4 + 4; ++ct) {
        const int colbase = ct * 16;
        v8f acc[5];
        const float bias = bv[colbase + m];
        #pragma unroll
        for (int rt = 0; rt < 5; ++rt)
            #pragma unroll
            for (int r = 0; r < 8; ++r) acc[rt][r] = bias;

        for (int kk = 0; kk < DD; kk += 4) {
            v2f bf;
            bf.x = Wm[(size_t)(kk + 2 * kg) * DD + colbase + m];
            bf.y = Wm[(size_t)(kk + 2 * kg + 1) * DD + colbase + m];
            #pragma unroll
            for (int rt = 0; rt < 5; ++rt) {
                v2f a;
                a.x = tile[rt * 16 + m][kk + 2 * kg];
                a.y = tile[rt * 16 + m][kk + 2 * kg + 1];
                acc[rt] = __builtin_amdgcn_wmma_f32_16x16x4_f32(
                    false, a, false, bf, (short)0, acc[rt], false, false);
            }
        }
        #pragma unroll
        for (int rt = 0; rt < 5; ++rt)
            #pragma unroll
            for (int r = 0; r < 8; ++r) {
                size_t row = row0 + rt * 16 + r + 8 * kg;
                outp[row * DD + colbase + m] = acc[rt][r];
            }
    }
}

// =====================================================================
// Kernel 2: per-edge fused  e_new = e@W2 + b2 + Dh[src] + Eh[dst]
//   sig = sigmoid(e_new); atomic num/den; write e_new*snorm_e to out.
// One block = 64 edges (4 row-tiles); 8 waves = 8 column tiles of 16.
// =====================================================================
__global__ __launch_bounds__(256) void edge_kernel(
    const float* __restrict__ e, const int* __restrict__ src,
    const int* __restrict__ dst, const float* __restrict__ W,
    const float* __restrict__ b, const float* __restrict__ Bh,
    const float* __restrict__ Dh, const float* __restrict__ Eh,
    const float* __restrict__ snorm_e,
    float* __restrict__ num, float* __restrict__ den,
    float* __restrict__ e_out)
{
    __shared__ float tile[ETILE][132];
    __shared__ int ssrc[ETILE], sdst[ETILE];
    const int t  = threadIdx.x;
    const size_t e0 = (size_t)blockIdx.x * ETILE;

    {
        const float* gbase = e + e0 * DD;
#if USE_ASYNC_LDS
        #pragma unroll
        for (int i = 0; i < 8; ++i) {
            int q = t + 256 * i;
            int row = q >> 5;
            int c4  = q & 31;
            __builtin_amdgcn_global_load_async_to_lds_b128(
                (GLOBAL_AS v4i*)(gbase + (size_t)row * DD + c4 * 4),
                (LDS_AS v4i*)&tile[row][c4 * 4], 0, 0);
        }
        __builtin_amdgcn_s_wait_asynccnt(0);
#else
        const float4* s = (const float4*)gbase;
        #pragma unroll
        for (int i = 0; i < 8; ++i) {
            int q = t + 256 * i;
            int row = q >> 5;
            int c4  = q & 31;
            float4 v = s[row * 32 + c4];
            *(float4*)&tile[row][c4 * 4] = v;
        }
#endif
    }
    if (t < ETILE) { ssrc[t] = src[e0 + t]; sdst[t] = dst[e0 + t]; }
    __syncthreads();

    const int lane = t & 31;
    const int colbase = (t >> 5) * 16;
    const float* __restrict__ W2 = W + (size_t)2 * DD * DD;
    const float sn = snorm_e[0];
    const int m  = lane & 15;
    const int kg = lane >> 4;

    v8f acc[4];
    const float bias = b[2 * DD + colbase + m];
    #pragma unroll
    for (int rt = 0; rt < 4; ++rt)
        #pragma unroll
        for (int r = 0; r < 8; ++r) acc[rt][r] = bias;

    for (int kk = 0; kk < DD; kk += 4) {
        v2f bf;
        bf.x = W2[(size_t)(kk + 2 * kg) * DD + colbase + m];
        bf.y = W2[(size_t)(kk + 2 * kg + 1) * DD + colbase + m];
        #pragma unroll
        for (int rt = 0; rt < 4; ++rt) {
            v2f a;
            a.x = tile[rt * 16 + m][kk + 2 * kg];
            a.y = tile[rt * 16 + m][kk + 2 * kg + 1];
            acc[rt] = __builtin_amdgcn_wmma_f32_16x16x4_f32(
                false, a, false, bf, (short)0, acc[rt], false, false);
        }
    }

    const int n = colbase + m;
    #pragma unroll
    for (int rt = 0; rt < 4; ++rt) {
        #pragma unroll
        for (int r = 0; r < 8; ++r) {
            const int erow = rt * 16 + r + 8 * kg;        // 0..63 in tile
            const size_t eidx = e0 + erow;
            const int s  = ssrc[erow];
            const int d0 = sdst[erow];
            float enew = acc[rt][r] + Dh[(size_t)s * DD + n] + Eh[(size_t)d0 * DD + n];
            float sg = 1.0f / (1.0f + expf(-enew));
            atomicAdd(&den[(size_t)d0 * DD + n], sg);
            atomicAdd(&num[(size_t)d0 * DD + n], sg * Bh[(size_t)s * DD + n]);
            e_out[eidx * DD + n] = enew * sn;             // pre-norm e, snorm applied
        }
    }
}

// ---------------- degree counts ----------------
__global__ void deg_kernel(const int* __restrict__ dst, float* __restrict__ deg)
{
    int i = blockIdx.x * blockDim.x + threadIdx.x;
    if (i < NE) atomicAdd(&deg[dst[i]], 1.0f);
}

// =====================================================================
// Kernel 3: h pre-norm + per-graph stats accumulation
// =====================================================================
__global__ __launch_bounds__(256) void h_pre_stats_kernel(
    const float* __restrict__ h, const float* __restrict__ Ah,
    const float* __restrict__ num, const float* __restrict__ den,
    const float* __restrict__ deg, const float* __restrict__ snorm_n,
    float* __restrict__ hout, float* __restrict__ hsum, float* __restrict__ hsum2)
{
    const int g = blockIdx.x / 20;
    const int chunk = blockIdx.x % 20;
    const int row0 = g * GNODE + chunk * 250;
    const int t = threadIdx.x;
    const int f = t & 127;
    const int half = t >> 7;
    const float sn = snorm_n[0];

    float s = 0.f, s2 = 0.f;
    for (int i = half; i < 250; i += 2) {
        const int row = row0 + i;
        const size_t idx = (size_t)row * DD + f;
        float v = (deg[row] > 0.f) ? (Ah[idx] + num[idx] / (den[idx] + EPS_RED))
                                   : h[idx];
        v *= sn;
        hout[idx] = v;
        s += v; s2 += v * v;
    }
    __shared__ float red[256];
    red[t] = s;  __syncthreads();
    if (half == 0) atomicAdd(&hsum[g * DD + f], red[f] + red[128 + f]);
    __syncthreads();
    red[t] = s2; __syncthreads();
    if (half == 0) atomicAdd(&hsum2[g * DD + f], red[f] + red[128 + f]);
}

// ---------------- e stats (e already staged in d_out) ----------------
__global__ __launch_bounds__(256) void e_stats_kernel(
    const float* __restrict__ eout, float* __restrict__ esum, float* __restrict__ esum2)
{
    const int g = blockIdx.x / 240;
    const int chunk = blockIdx.x % 240;
    const int row0 = g * GEDGE + chunk * 250;
    const int t = threadIdx.x;
    const int f = t & 127;
    const int half = t >> 7;

    float s = 0.f, s2 = 0.f;
    for (int i = half; i < 250; i += 2) {
        const size_t idx = (size_t)(row0 + i) * DD + f;
        float v = eout[idx];
        s += v; s2 += v * v;
    }
    __shared__ float red[256];
    red[t] = s;  __syncthreads();
    if (half == 0) atomicAdd(&esum[g * DD + f], red[f] + red[128 + f]);
    __syncthreads();
    red[t] = s2; __syncthreads();
    if (half == 0) atomicAdd(&esum2[g * DD + f], red[f] + red[128 + f]);
}

// ---------------- finalize: mean & 1/(std+eps), ddof=1 ----------------
__global__ void finalize_stats_kernel(
    const float* __restrict__ hsum, const float* __restrict__ hsum2,
    const float* __restrict__ esum, const float* __restrict__ esum2,
    float* __restrict__ hmean, float* __restrict__ hinv,
    float* __restrict__ emean, float* __restrict__ einv)
{
    int i = blockIdx.x * blockDim.x + threadIdx.x;
    if (i < 1280) {
        const float sz = (float)GNODE;
        float mean = hsum[i] / sz;
        float var = (hsum2[i] - sz * mean * mean) / (sz - 1.f);
        var = var < 0.f ? 0.f : var;
        hmean[i] = mean;
        hinv[i]  = 1.f / (sqrtf(var) + EPS_NORM);
    } else if (i < 2560) {
        int j = i - 1280;
        const float sz = (float)GEDGE;
        float mean = esum[j] / sz;
        float var = (esum2[j] - sz * mean * mean) / (sz - 1.f);
        var = var < 0.f ? 0.f : var;
        emean[j] = mean;
        einv[j]  = 1.f / (sqrtf(var) + EPS_NORM);
    }
}

// ---------------- in-place normalize + relu + residual ----------------
__global__ __launch_bounds__(256) void h_final_kernel(
    const float* __restrict__ h_in, const float* __restrict__ gamma,
    const float* __restrict__ beta, const float* __restrict__ mean,
    const float* __restrict__ inv, float* __restrict__ hout)
{
    size_t i = (size_t)blockIdx.x * 256 + threadIdx.x;   // exact: ND
    int f = (int)(i & 127);
    int row = (int)(i >> 7);
    int g = row / GNODE;
    float v = hout[i];
    float o = gamma[f] * ((v - mean[g * DD + f]) * inv[g * DD + f]) + beta[f];
    o = fmaxf(o, 0.f);
    hout[i] = h_in[i] + o;
}

__global__ __launch_bounds__(256) void e_final_kernel(
    const float* __restrict__ e_in, const float* __restrict__ gamma,
    const float* __restrict__ beta, const float* __restrict__ mean,
    const float* __restrict__ inv, float* __restrict__ eout)
{
    size_t i = (size_t)blockIdx.x * 256 + threadIdx.x;   // exact: ED
    int f = (int)(i & 127);
    int row = (int)(i >> 7);
    int g = row / GEDGE;
    float v = eout[i];
    float o = gamma[f] * ((v - mean[g * DD + f]) * inv[g * DD + f]) + beta[f];
    o = fmaxf(o, 0.f);
    eout[i] = e_in[i] + o;
}

// =====================================================================
extern "C" void kernel_launch(void* const* d_in, const int* in_sizes, int n_in,
                              void* d_out, int out_size, void* d_ws, size_t ws_size,
                              hipStream_t stream)
{
    const float* h   = (const float*)d_in[0];
    const float* e   = (const float*)d_in[1];
    const int*   src = (const int*)d_in[2];
    const int*   dst = (const int*)d_in[3];
    const float* snn = (const float*)d_in[4];
    const float* sne = (const float*)d_in[5];
    const float* W   = (const float*)d_in[6];
    const float* b   = (const float*)d_in[7];
    const float* gh  = (const float*)d_in[8];
    const float* bh  = (const float*)d_in[9];
    const float* ge  = (const float*)d_in[10];
    const float* be  = (const float*)d_in[11];

    float* out_h = (float*)d_out;
    float* out_e = out_h + ND;

    // workspace layout (floats)
    float* ws   = (float*)d_ws;
    float* Ah   = ws + 0 * ND;
    float* Bh   = ws + 1 * ND;
    float* Dh   = ws + 2 * ND;
    float* Eh   = ws + 3 * ND;
    float* num  = ws + 4 * ND;
    float* den  = ws + 5 * ND;
    float* deg  = ws + 6 * ND;
    float* hsum  = deg + NN;
    float* hsum2 = hsum  + 1280;
    float* esum  = hsum2 + 1280;
    float* esum2 = esum  + 1280;
    float* hmean = esum2 + 1280;
    float* hinv  = hmean + 1280;
    float* emean = hinv  + 1280;
    float* einv  = emean + 1280;

    // zero accumulators: num, den, deg, 4 stats arrays (contiguous range)
    size_t zero_floats = 2 * ND + NN + 4 * 1280;
    (void)hipMemsetAsync(num, 0, zero_floats * sizeof(float), stream);

    node_gemm_kernel<<<NN / NTILE, 256, 0, stream>>>(h, W, b, Ah, Bh, Dh, Eh);
    deg_kernel<<<(NE + 255) / 256, 256, 0, stream>>>(dst, deg);
    edge_kernel<<<NE / ETILE, 256, 0, stream>>>(e, src, dst, W, b, Bh, Dh, Eh,
                                                sne, num, den, out_e);
    h_pre_stats_kernel<<<10 * 20, 256, 0, stream>>>(h, Ah, num, den, deg, snn,
                                                    out_h, hsum, hsum2);
    e_stats_kernel<<<10 * 240, 256, 0, stream>>>(out_e, esum, esum2);
    finalize_stats_kernel<<<10, 256, 0, stream>>>(hsum, hsum2, esum, esum2,
                                                  hmean, hinv, emean, einv);
    h_final_kernel<<<(unsigned)(ND / 256), 256, 0, stream>>>(h, gh, bh, hmean, hinv, out_h);
    e_final_kernel<<<(unsigned)(ED / 256), 256, 0, stream>>>(e, ge, be, emean, einv, out_e);
}